// CorrelationLayer1D_24945170055765
// MI455X (gfx1250) — compile-verified
//
#include <hip/hip_runtime.h>

// CDNA5 / gfx1250: fp32 WMMA (16x16x4) correlation layer.
// out[b,d,h,w] = sum_c x1[b,c,h,w] * x2[b,c,h,w+d-20],  D = 41 shifts.
//
// Mapping: for fixed (b,h),  Out[w,d] = sum_c A[w,c] * Bp[c, q],  q = w+d  (Bp zero-padded x2).
// Tile w in 16s; compute P[w0+i, q0+n] for 4 q-tiles (q0 = w0 + 16t), then scatter
// P -> Out[w, q-w].  Only waves whose *used* columns touch the padding need masking.

typedef float v2f __attribute__((ext_vector_type(2)));
typedef float v8f __attribute__((ext_vector_type(8)));

#define B_     8
#define C_     256
#define H_     96
#define W_     320
#define D_     41
#define PAD    20
#define WAVES  4      // waves per block, each owns a 16-wide w tile
#define GROUPS 5      // (W/16)/WAVES = 20/4

template <bool MASKED>
__device__ __forceinline__ void corr_loop(const float* __restrict__ aPtr,
                                          const float* __restrict__ bPtr0,
                                          const float* __restrict__ bPtr1,
                                          const float* __restrict__ bPtr2,
                                          const float* __restrict__ bPtr3,
                                          float ok0, float ok1, float ok2, float ok3,
                                          v8f& acc0, v8f& acc1, v8f& acc2, v8f& acc3) {
    const size_t cs = (size_t)H_ * W_;   // channel stride

    #pragma unroll 8
    for (int c = 0; c < C_; c += 4) {
        v2f a;
        a.x = aPtr[0];
        a.y = aPtr[cs];

        v2f b0, b1, b2, b3;
        if (MASKED) {
            b0.x = bPtr0[0] * ok0;  b0.y = bPtr0[cs] * ok0;
            b1.x = bPtr1[0] * ok1;  b1.y = bPtr1[cs] * ok1;
            b2.x = bPtr2[0] * ok2;  b2.y = bPtr2[cs] * ok2;
            b3.x = bPtr3[0] * ok3;  b3.y = bPtr3[cs] * ok3;
        } else {
            b0.x = bPtr0[0];  b0.y = bPtr0[cs];
            b1.x = bPtr1[0];  b1.y = bPtr1[cs];
            b2.x = bPtr2[0];  b2.y = bPtr2[cs];
            b3.x = bPtr3[0];  b3.y = bPtr3[cs];
        }

        aPtr  += 4 * cs;
        bPtr0 += 4 * cs;
        bPtr1 += 4 * cs;
        bPtr2 += 4 * cs;
        bPtr3 += 4 * cs;

        // D = A x B + C   (8 args: neg_a, A, neg_b, B, c_mod, C, reuse_a, reuse_b)
        acc0 = __builtin_amdgcn_wmma_f32_16x16x4_f32(false, a, false, b0, (short)0, acc0, false, false);
        acc1 = __builtin_amdgcn_wmma_f32_16x16x4_f32(false, a, false, b1, (short)0, acc1, false, false);
        acc2 = __builtin_amdgcn_wmma_f32_16x16x4_f32(false, a, false, b2, (short)0, acc2, false, false);
        acc3 = __builtin_amdgcn_wmma_f32_16x16x4_f32(false, a, false, b3, (short)0, acc3, false, false);
    }
}

__global__ __launch_bounds__(32 * WAVES)
void corr1d_wmma_f32_kernel(const float* __restrict__ x1,
                            const float* __restrict__ x2,
                            float* __restrict__ out) {
    const int lane = threadIdx.x & 31;
    const int wave = threadIdx.x >> 5;

    int gid = blockIdx.x;
    const int group = gid % GROUPS;  gid /= GROUPS;
    const int h     = gid % H_;
    const int b     = gid / H_;

    const int w0 = (group * WAVES + wave) * 16;

    // WMMA f32 16x16x4 register mapping (wave32):
    //  A (16x4): lane m = lane&15 holds row M=m; half-wave selects K pair:
    //            VGPR0 = K = k0, VGPR1 = K = k0+1, k0 = (lane>>4)*2.
    //  B (4x16): lane n = lane&15 holds col N=n; same K striping as A.
    const int m  = lane & 15;
    const int k0 = (lane >> 4) << 1;

    const size_t cs = (size_t)H_ * W_;

    const float* x1p = x1 + ((size_t)b * C_ * H_ + h) * W_;   // [C][W] slab
    const float* x2p = x2 + ((size_t)b * C_ * H_ + h) * W_;   // [C][W] slab

    const float* aPtr = x1p + (size_t)k0 * cs + (size_t)(w0 + m);

    // B columns: q = w0 + 16*t + n, source col = q - PAD; clamp OOB addresses.
    const int qs0 = w0 +  0 + m - PAD;
    const int qs1 = w0 + 16 + m - PAD;
    const int qs2 = w0 + 32 + m - PAD;
    const int qs3 = w0 + 48 + m - PAD;
    const float ok0 = (qs0 >= 0 && qs0 < W_) ? 1.0f : 0.0f;
    const float ok1 = (qs1 >= 0 && qs1 < W_) ? 1.0f : 0.0f;
    const float ok2 = (qs2 >= 0 && qs2 < W_) ? 1.0f : 0.0f;
    const float ok3 = (qs3 >= 0 && qs3 < W_) ? 1.0f : 0.0f;
    const float* bPtr0 = x2p + (size_t)k0 * cs + (size_t)((ok0 != 0.0f) ? qs0 : 0);
    const float* bPtr1 = x2p + (size_t)k0 * cs + (size_t)((ok1 != 0.0f) ? qs1 : 0);
    const float* bPtr2 = x2p + (size_t)k0 * cs + (size_t)((ok2 != 0.0f) ? qs2 : 0);
    const float* bPtr3 = x2p + (size_t)k0 * cs + (size_t)((ok3 != 0.0f) ? qs3 : 0);

    v8f acc0 = {}, acc1 = {}, acc2 = {}, acc3 = {};

    // Masking is required only if a *used* column (q <= w0+55) falls in the padding:
    //   low edge:  w0 < PAD        (w0 = 0, 16)
    //   high edge: w0 + 55 - PAD >= W_   i.e. w0 > 284  (w0 = 288, 304)
    // This predicate is wave-uniform, so the branch is free of divergence.
    const bool needMask = (w0 < PAD) || (w0 + 55 - PAD >= W_);

    if (needMask) {
        corr_loop<true>(aPtr, bPtr0, bPtr1, bPtr2, bPtr3,
                        ok0, ok1, ok2, ok3, acc0, acc1, acc2, acc3);
    } else {
        corr_loop<false>(aPtr, bPtr0, bPtr1, bPtr2, bPtr3,
                         ok0, ok1, ok2, ok3, acc0, acc1, acc2, acc3);
    }

    // C/D layout: VGPR r, lanes 0-15 -> row i=r, lanes 16-31 -> row i=r+8; col n=lane&15.
    // Scatter P[w0+i, q=w0+16t+n] into out[b, d=16t+n-i, h, w0+i].
    const int iBase = (lane >> 4) << 3;
    float* outp = out + ((size_t)b * D_ * H_ + h) * W_ + (size_t)w0;

    #pragma unroll
    for (int r = 0; r < 8; ++r) {
        const int i = iBase + r;
        {
            const int d = 0 * 16 + m - i;
            if (d >= 0 && d < D_) outp[(size_t)d * cs + i] = acc0[r];
        }
        {
            const int d = 1 * 16 + m - i;
            if (d >= 0 && d < D_) outp[(size_t)d * cs + i] = acc1[r];
        }
        {
            const int d = 2 * 16 + m - i;
            if (d >= 0 && d < D_) outp[(size_t)d * cs + i] = acc2[r];
        }
        {
            const int d = 3 * 16 + m - i;
            if (d >= 0 && d < D_) outp[(size_t)d * cs + i] = acc3[r];
        }
    }
}

extern "C" void kernel_launch(void* const* d_in, const int* in_sizes, int n_in,
                              void* d_out, int out_size, void* d_ws, size_t ws_size,
                              hipStream_t stream) {
    (void)in_sizes; (void)n_in; (void)out_size; (void)d_ws; (void)ws_size;
    const float* x1 = (const float*)d_in[0];
    const float* x2 = (const float*)d_in[1];
    float* out = (float*)d_out;

    const int numBlocks = B_ * H_ * GROUPS;   // 8 * 96 * 5 = 3840
    corr1d_wmma_f32_kernel<<<numBlocks, 32 * WAVES, 0, stream>>>(x1, x2, out);
}